// GNNModel_9569187135821
// MI455X (gfx1250) — compile-verified
//
#include <hip/hip_runtime.h>
#include <hip/hip_bf16.h>

typedef __attribute__((ext_vector_type(2))) float v2f;
typedef __attribute__((ext_vector_type(8))) float v8f;

#define A_STRIDE 132  // floats: 528B rows -> 16B-aligned, conflict-free ds_load_b64

// ---------------------------------------------------------------- utilities
__global__ void zero_f32_kernel(float* __restrict__ p, long n) {
  long i = (long)blockIdx.x * blockDim.x + threadIdx.x;
  long stride = (long)gridDim.x * blockDim.x;
  for (; i < n; i += stride) p[i] = 0.0f;
}

// CDNA5 async global->LDS copy of one 16B chunk (tracked by ASYNCcnt)
__device__ __forceinline__ void async_cp16(unsigned lds_off, const float* gaddr) {
  asm volatile("global_load_async_to_lds_b128 %0, %1, off"
               :: "v"(lds_off), "v"(gaddr)
               : "memory");
}

__device__ __forceinline__ void wait_async0() {
#if __has_builtin(__builtin_amdgcn_s_wait_asynccnt)
  __builtin_amdgcn_s_wait_asynccnt(0);
#else
  asm volatile("s_wait_asynccnt 0x0" ::: "memory");
#endif
}

// ------------------------------------------------- layer 1: edge scatter (d=3)
__global__ void scatter_x_kernel(const float* __restrict__ x,
                                 const int* __restrict__ src,
                                 const int* __restrict__ dst,
                                 float* agg1, int n_edges) {
  int e = blockIdx.x * blockDim.x + threadIdx.x;
  if (e >= n_edges) return;
  int s = src[e], d = dst[e];
  const float* xs = x + (long)s * 3;
  float* ad = agg1 + (long)d * 3;
  atomicAdd(ad + 0, xs[0]);
  atomicAdd(ad + 1, xs[1]);
  atomicAdd(ad + 2, xs[2]);
}

// -------------------------- layer 1 dense part: h1 = relu(agg@W1r + b1 + x@W1t)
__global__ void layer1_kernel(const float* __restrict__ x,
                              const float* __restrict__ agg1,
                              const float* __restrict__ W_rel,   // [3,128]
                              const float* __restrict__ b1,      // [128]
                              const float* __restrict__ W_root,  // [3,128]
                              float* __restrict__ h1, int n_nodes) {
  int node = blockIdx.x;
  if (node >= n_nodes) return;
  int k = threadIdx.x;  // 0..127
  float a0 = agg1[(long)node * 3 + 0];
  float a1 = agg1[(long)node * 3 + 1];
  float a2 = agg1[(long)node * 3 + 2];
  float x0 = x[(long)node * 3 + 0];
  float x1 = x[(long)node * 3 + 1];
  float x2 = x[(long)node * 3 + 2];
  float v = b1[k];
  v = fmaf(a0, W_rel[k], v);
  v = fmaf(a1, W_rel[128 + k], v);
  v = fmaf(a2, W_rel[256 + k], v);
  v = fmaf(x0, W_root[k], v);
  v = fmaf(x1, W_root[128 + k], v);
  v = fmaf(x2, W_root[256 + k], v);
  h1[(long)node * 128 + k] = fmaxf(v, 0.0f);
}

// ----------------------------------- layer 2: edge scatter of 128-float rows
__global__ void scatter_h1_kernel(const float* __restrict__ h1,
                                  const int* __restrict__ src,
                                  const int* __restrict__ dst,
                                  float* agg2, int n_edges) {
  long tid = (long)blockIdx.x * blockDim.x + threadIdx.x;
  int e = (int)(tid >> 5);
  int lane = (int)(tid & 31);
  if (e >= n_edges) return;
  int s = src[e], d = dst[e];
  const float4 v = *(const float4*)(h1 + (long)s * 128 + lane * 4);
  float* base = agg2 + (long)d * 128 + lane * 4;
  atomicAdd(base + 0, v.x);
  atomicAdd(base + 1, v.y);
  atomicAdd(base + 2, v.z);
  atomicAdd(base + 3, v.w);
}

// --------- layer 2 dense (WMMA): h2 = relu(agg2@W2r + b2 + h1@W2t), in-place.
// Persistent blocks: 8 waves; wave w owns output cols [16w,16w+16) and keeps
// its B fragments of BOTH weight matrices in 128 VGPRs for the whole kernel.
// A tiles (16 rows x 128 of agg2 and h1) are double-buffered in LDS via
// global_load_async_to_lds_b128 so the copy of tile t+1 overlaps the 64
// chained v_wmma_f32_16x16x4_f32 of tile t.
__global__ __launch_bounds__(256) void layer2_wmma_kernel(
    float* agg2_h2,                    // [n,128] input aggregate / output h2
    const float* __restrict__ h1,      // [n,128]
    const float* __restrict__ W_rel,   // [128,128]
    const float* __restrict__ b2v,     // [128]
    const float* __restrict__ W_root,  // [128,128]
    int n_nodes, int n_tiles) {
  __shared__ float sA[2][2][16 * A_STRIDE];  // [buf][matrix] : 33 KB total

  const int tid  = threadIdx.x;
  const int lane = tid & 31;
  const int wave = tid >> 5;   // 0..7
  const int half = lane >> 4;  // 0|1  (K sub-pair for 16x16x4 f32 layout)
  const int l16  = lane & 15;
  const int tile_n = wave * 16;
  const int col = tile_n + l16;

  // ---- preload B fragments of both weight matrices into registers
  v2f Brel[32], Brt[32];
#pragma unroll
  for (int kk = 0; kk < 32; ++kk) {
    const int ka = kk * 4 + half * 2;
    Brel[kk].x = W_rel[ka * 128 + col];
    Brel[kk].y = W_rel[(ka + 1) * 128 + col];
    Brt[kk].x  = W_root[ka * 128 + col];
    Brt[kk].y  = W_root[(ka + 1) * 128 + col];
  }
  const float bias = b2v[col];

  // async-copy coordinates: per tile each thread moves 4 x 16B
  const int crow = tid >> 5;  // base row 0..7 (second pass adds 8)
  const int cq   = tid & 31;  // 16B chunk within a 512B row

  const unsigned lds_base[2] = {(unsigned)(uintptr_t)&sA[0][0][0],
                                (unsigned)(uintptr_t)&sA[1][0][0]};

  int tile = blockIdx.x;
  int cur = 0;

  // prologue: kick off first tile's copies
  if (tile < n_tiles) {
#pragma unroll
    for (int m = 0; m < 2; ++m) {
      const float* srcm = (m == 0) ? agg2_h2 : h1;
      const unsigned mbase = lds_base[0] + (unsigned)(m * 16 * A_STRIDE * 4);
#pragma unroll
      for (int p = 0; p < 2; ++p) {
        const int r = crow + p * 8;
        const int row = tile * 16 + r;
        if (row < n_nodes)
          async_cp16(mbase + (unsigned)(r * A_STRIDE * 4 + cq * 16),
                     srcm + (long)row * 128 + cq * 4);
      }
    }
  }

  for (; tile < n_tiles; tile += gridDim.x) {
    wait_async0();     // this wave's chunks landed in LDS
    __syncthreads();   // everyone's chunks landed; prev buffer free for reuse

    // kick off next tile into the other buffer (overlaps compute below)
    const int nxt = tile + gridDim.x;
    if (nxt < n_tiles) {
#pragma unroll
      for (int m = 0; m < 2; ++m) {
        const float* srcm = (m == 0) ? agg2_h2 : h1;
        const unsigned mbase =
            lds_base[cur ^ 1] + (unsigned)(m * 16 * A_STRIDE * 4);
#pragma unroll
        for (int p = 0; p < 2; ++p) {
          const int r = crow + p * 8;
          const int row = nxt * 16 + r;
          if (row < n_nodes)
            async_cp16(mbase + (unsigned)(r * A_STRIDE * 4 + cq * 16),
                       srcm + (long)row * 128 + cq * 4);
        }
      }
    }

    // ---- 64 chained f32 WMMAs from the LDS snapshot
    const float* aRel = &sA[cur][0][l16 * A_STRIDE + half * 2];
    const float* aRt  = &sA[cur][1][l16 * A_STRIDE + half * 2];
    v8f acc = {};
#pragma unroll
    for (int kk = 0; kk < 32; ++kk) {
      v2f a = *(const v2f*)(aRel + kk * 4);
      acc = __builtin_amdgcn_wmma_f32_16x16x4_f32(false, a, false, Brel[kk],
                                                  (short)0, acc, false, false);
      v2f a2 = *(const v2f*)(aRt + kk * 4);
      acc = __builtin_amdgcn_wmma_f32_16x16x4_f32(false, a2, false, Brt[kk],
                                                  (short)0, acc, false, false);
    }

    // store (in-place over agg2 rows; safe — compute used the LDS snapshot)
#pragma unroll
    for (int r = 0; r < 8; ++r) {
      const int m = tile * 16 + r + half * 8;  // lanes>=16 hold M=r+8
      if (m < n_nodes) {
        float v = acc[r] + bias;
        agg2_h2[(long)m * 128 + tile_n + l16] = fmaxf(v, 0.0f);
      }
    }
    cur ^= 1;
  }
}

// ----------------------------------------------- mean-pool scatter by graph id
__global__ void pool_kernel(const float* __restrict__ h2,
                            const int* __restrict__ batch,
                            float* sums, float* counts, int n_nodes) {
  long tid = (long)blockIdx.x * blockDim.x + threadIdx.x;
  int node = (int)(tid >> 7);
  int k = (int)(tid & 127);
  if (node >= n_nodes) return;
  int g = batch[node];
  atomicAdd(&sums[(long)g * 128 + k], h2[(long)node * 128 + k]);
  if (k == 0) atomicAdd(&counts[g], 1.0f);
}

// ------------------------------------------ FC head: sigmoid(mean @ Wfc + bfc)
__global__ void fc_kernel(const float* __restrict__ sums,
                          const float* __restrict__ counts,
                          const float* __restrict__ Wfc,  // [128,138]
                          const float* __restrict__ bfc,  // [138]
                          float* __restrict__ out, int n_graphs) {
  long tid = (long)blockIdx.x * blockDim.x + threadIdx.x;
  int g = (int)(tid / 138);
  int k = (int)(tid % 138);
  if (g >= n_graphs) return;
  float inv = 1.0f / fmaxf(counts[g], 1.0f);
  const float* s = sums + (long)g * 128;
  float acc = bfc[k];
#pragma unroll 8
  for (int j = 0; j < 128; ++j) acc = fmaf(s[j] * inv, Wfc[j * 138 + k], acc);
  out[(long)g * 138 + k] = 1.0f / (1.0f + expf(-acc));
}

// --------------------------------------------------------------------- launch
extern "C" void kernel_launch(void* const* d_in, const int* in_sizes, int n_in,
                              void* d_out, int out_size, void* d_ws, size_t ws_size,
                              hipStream_t stream) {
  const float* x      = (const float*)d_in[0];
  const int*   eidx   = (const int*)d_in[1];    // [2, E] flat
  const int*   batch  = (const int*)d_in[2];
  const float* W1_rel = (const float*)d_in[4];
  const float* b1     = (const float*)d_in[5];
  const float* W1_root= (const float*)d_in[6];
  const float* W2_rel = (const float*)d_in[7];
  const float* b2     = (const float*)d_in[8];
  const float* W2_root= (const float*)d_in[9];
  const float* Wfc    = (const float*)d_in[10];
  const float* bfc    = (const float*)d_in[11];
  float* out = (float*)d_out;

  const int n_nodes  = in_sizes[0] / 3;
  const int n_edges  = in_sizes[1] / 2;
  const int n_graphs = out_size / 138;
  const int* src = eidx;
  const int* dst = eidx + n_edges;

  // workspace layout (floats): [agg1][agg2/h2][sums][counts][h1]
  float* ws   = (float*)d_ws;
  float* agg1 = ws;
  float* agg2 = agg1 + (long)n_nodes * 3;
  float* sums = agg2 + (long)n_nodes * 128;
  float* cnts = sums + (long)n_graphs * 128;
  float* h1   = cnts + (long)n_graphs;
  const long zero_span = (long)(h1 - agg1);

  // 1) zero accumulator buffers (agg1, agg2, sums, counts are contiguous)
  zero_f32_kernel<<<2048, 256, 0, stream>>>(agg1, zero_span);

  // 2) layer-1 edge scatter (d=3)
  scatter_x_kernel<<<(n_edges + 255) / 256, 256, 0, stream>>>(x, src, dst, agg1, n_edges);

  // 3) layer-1 dense
  layer1_kernel<<<n_nodes, 128, 0, stream>>>(x, agg1, W1_rel, b1, W1_root, h1, n_nodes);

  // 4) layer-2 edge scatter
  {
    long threads = (long)n_edges * 32;
    scatter_h1_kernel<<<(unsigned)((threads + 255) / 256), 256, 0, stream>>>(
        h1, src, dst, agg2, n_edges);
  }

  // 5) layer-2 dense via persistent f32-WMMA kernel (in-place agg2 -> h2)
  {
    const int n_tiles = (n_nodes + 15) / 16;
    int blocks = n_tiles < 1920 ? n_tiles : 1920;
    layer2_wmma_kernel<<<blocks, 256, 0, stream>>>(
        agg2, h1, W2_rel, b2, W2_root, n_nodes, n_tiles);
  }

  // 6) mean-pool scatter
  {
    long threads = (long)n_nodes * 128;
    pool_kernel<<<(unsigned)((threads + 255) / 256), 256, 0, stream>>>(
        agg2, batch, sums, cnts, n_nodes);
  }

  // 7) FC + sigmoid
  {
    long threads = (long)n_graphs * 138;
    fc_kernel<<<(unsigned)((threads + 255) / 256), 256, 0, stream>>>(
        sums, cnts, Wfc, bfc, out, n_graphs);
  }
}